// RoIPooling_64974265254484
// MI455X (gfx1250) — compile-verified
//
#include <hip/hip_runtime.h>
#include <hip/hip_bf16.h>

// Problem constants (from reference)
#define BATCH   4
#define RBOX    128
#define NREG    32
#define FH      64
#define FW      64
#define FC      256
#define PH      3
#define PW      3
#define IOU_THR 0.4f
#define NEG_INF (-3.402823466e+38f)
#define CHUNK_PX 16   // pixels per staged chunk (16 px * 256 ch * 4B = 16 KB)

typedef unsigned int u32x4 __attribute__((ext_vector_type(4)));
typedef int          i32x4 __attribute__((ext_vector_type(4)));
typedef int          i32x8 __attribute__((ext_vector_type(8)));

// ---------------------------------------------------------------------------
// Python floor-div by 2 (arithmetic shift == floor for clang)
__device__ __forceinline__ int fdiv2(int a) { return a >> 1; }

// Reference fix_axis, integer math, Python floor-div semantics.
__device__ __forceinline__ void fix_axis(int& mn, int& mx, int ps, int fs) {
    int pad = ps - (mx - mn);
    bool fix_min = mn < fdiv2(pad);
    bool fix_max = (fs - mx) < fdiv2(1 + pad);
    bool sym = (pad > 0) && !(fix_min || fix_max);
    int omin = sym ? (mn - fdiv2(pad))     : mn;
    int omax = sym ? (mx + fdiv2(1 + pad)) : mx;
    if (pad > 0 && fix_min) { omin = 0;       omax = ps; }
    if (pad > 0 && fix_max) { omin = fs - ps; omax = fs; }   // fix_max overrides (reference order)
    mn = omin; mx = omax;
}

// ---------------------------------------------------------------------------
// Kernel 1: per-batch pairwise IoU + sequential NMS + gather + clip.
// Grid: BATCH blocks, 128 threads (thread t owns box t).
__global__ void nms_clip_kernel(const float* __restrict__ roi,
                                int*   __restrict__ boxes_ws,
                                float* __restrict__ roi_out) {
    __shared__ float bx[RBOX], by[RBOX], bw[RBOX], bh[RBOX];
    __shared__ int   keep[RBOX];
    __shared__ int   sel[NREG];

    const int b = blockIdx.x;
    const int t = threadIdx.x;

    const float* rb = roi + (size_t)(b * RBOX + t) * 4;
    bx[t] = rb[0]; by[t] = rb[1]; bw[t] = rb[2]; bh[t] = rb[3];
    keep[t] = 1;
    __syncthreads();

    const float x1 = bx[t], y1 = by[t];
    const float x2 = x1 + bw[t], y2 = y1 + bh[t];
    const float area = bw[t] * bh[t];

    // Sequential suppression, mirroring the reference fori_loop:
    // at step p, suppress j>p with iou(p,j) > thr if keep[p] currently set.
    for (int p = 0; p < RBOX - 1; ++p) {
        __syncthreads();
        if (t > p && keep[t] && keep[p]) {
            float px1 = bx[p], py1 = by[p];
            float px2 = px1 + bw[p], py2 = py1 + bh[p];
            float parea = bw[p] * bh[p];
            float iw = fminf(px2, x2) - fmaxf(px1, x1); iw = fmaxf(iw, 0.0f);
            float ih = fminf(py2, y2) - fmaxf(py1, y1); ih = fmaxf(ih, 0.0f);
            float inter = iw * ih;
            float iou = inter / (parea + area - inter);
            if (iou > IOU_THR) keep[t] = 0;
        }
    }
    __syncthreads();

    // sort(where(keep, j, R))[:NREG], min(idx, R-1): ascending kept indices, pad 127.
    if (t == 0) {
        int n = 0;
        for (int j = 0; j < RBOX && n < NREG; ++j)
            if (keep[j]) sel[n++] = j;
        for (; n < NREG; ++n) sel[n] = RBOX - 1;
    }
    __syncthreads();

    if (t < NREG) {
        int s = sel[t];
        float x = bx[s], y = by[s], w = bw[s], h = bh[s];
        int xmin = (int)fmaxf(0.0f, x);
        int ymin = (int)fmaxf(0.0f, y);
        int xmax = (int)fminf((float)FW, x + w);
        int ymax = (int)fminf((float)FH, y + h);
        fix_axis(xmin, xmax, PW, FW);
        fix_axis(ymin, ymax, PH, FH);
        int ow = xmax - xmin, oh = ymax - ymin;

        int*   o = boxes_ws + (size_t)(b * NREG + t) * 4;
        float* f = roi_out  + (size_t)(b * NREG + t) * 4;
        o[0] = xmin; o[1] = ymin; o[2] = ow; o[3] = oh;
        f[0] = (float)xmin; f[1] = (float)ymin; f[2] = (float)ow; f[3] = (float)oh;
    }
}

// ---------------------------------------------------------------------------
// Issue one TDM descriptor: 1-row tile of `nelem` fp32 elements,
// contiguous Global -> LDS. nelem <= 4096 (fits tile_dim0 16-bit field).
// D# built per CDNA5 ISA 8.3/8.4 bit layout. Trailing groups zero (<=2D).
__device__ __forceinline__ void tdm_load_row_chunk(const float* __restrict__ gsrc,
                                                   unsigned lds_addr, int nelem) {
    unsigned long long ga = (unsigned long long)(const void*)gsrc;
    u32x4 g0;
    g0.x = 0x1u;                                          // count=1, user D#
    g0.y = lds_addr;                                      // LDS byte address
    g0.z = (unsigned)(ga & 0xFFFFFFFFull);                // global_addr[31:0]
    g0.w = (unsigned)((ga >> 32) & 0x01FFFFFFull)         // global_addr[56:32]
         | (2u << 30);                                    // type = 2 ("image")
    i32x8 g1;
    unsigned ne = (unsigned)nelem;
    g1.s0 = (int)(2u << 16);             // data_size=4B; mask/flags/pad = 0
    g1.s1 = (int)((ne & 0xFFFFu) << 16); // tensor_dim0[15:0] (abar addr = 0)
    g1.s2 = (int)(((ne >> 16) & 0xFFFFu) | (1u << 16)); // tdim0[31:16] | tensor_dim1=1
    g1.s3 = (int)((ne & 0xFFFFu) << 16); // tensor_dim1[31:16]=0 | tile_dim0=nelem
    g1.s4 = 1;                           // tile_dim1=1, tile_dim2=0
    g1.s5 = (int)ne;                     // tensor_dim0_stride[31:0]
    g1.s6 = 0;                           // stride0[47:32]=0, stride1[15:0]=0
    g1.s7 = 0;                           // stride1[47:16]=0
    i32x4 z4 = (i32x4){0, 0, 0, 0};
    i32x8 z8 = (i32x8){0, 0, 0, 0, 0, 0, 0, 0};
    // 6-arg form (clang-23 / therock-10.0 toolchain)
    __builtin_amdgcn_tensor_load_to_lds(g0, g1, z4, z4, z8, 0);
}

// ---------------------------------------------------------------------------
// Kernel 2: 3x3 max-pool of one clipped region per block.
// Grid: BATCH*NREG blocks, 256 threads (thread t = channel t).
// TDM (tensor_load_to_lds) double-buffered row-strip staging, overlapped with
// the channel-wise max reduction; boxes fetched via async global->LDS.
__global__ void pool_kernel(const float* __restrict__ features,
                            const int*  __restrict__ boxes,
                            float*      __restrict__ pooled) {
    __shared__ float sbuf[2][CHUNK_PX * FC];   // 2 x 16 KB
    __shared__ int   sbox[4];

    const int blk = blockIdx.x;
    const int b   = blk / NREG;
    const int t   = threadIdx.x;

    // CDNA5 async global->LDS path for the tiny box fetch (ASYNCcnt).
    if (t < 4) {
        unsigned long long ga =
            (unsigned long long)(const void*)(boxes + (size_t)blk * 4 + t);
        unsigned la = (unsigned)(size_t)(const void*)&sbox[t];
        asm volatile("global_load_async_to_lds_b32 %0, %1, off"
                     :: "v"(la), "v"(ga) : "memory");
    }
    asm volatile("s_wait_asynccnt 0x0" ::: "memory");
    __syncthreads();

    const int x = sbox[0], y = sbox[1], w = sbox[2], h = sbox[3];
    const int dw = w / PW;            // >= 1 (w >= 3 after clip)
    const int dh = h / PH;            // >= 1

    const float* fb = features + (size_t)b * FH * FW * FC;
    const int nchunks = (w + CHUNK_PX - 1) / CHUNK_PX;

    const unsigned lds0 = (unsigned)(size_t)(const void*)&sbuf[0][0];
    const unsigned lds1 = (unsigned)(size_t)(const void*)&sbuf[1][0];

    for (int rc = 0; rc < PH; ++rc) {
        const int rbeg  = y + rc * dh;
        const int rend  = (rc == PH - 1) ? (y + h) : (y + (rc + 1) * dh);
        const int total = (rend - rbeg) * nchunks;

        float m0 = NEG_INF, m1 = NEG_INF, m2 = NEG_INF;

        // Wave 0 drives the TDM engine (TDM ignores EXEC; branch is wave-uniform).
        if (t < 32) {
            int npx0 = (w < CHUNK_PX) ? w : CHUNK_PX;
            tdm_load_row_chunk(fb + ((size_t)rbeg * FW + x) * FC, lds0, npx0 * FC);
        }

        for (int k = 0; k < total; ++k) {
            const int cur = k & 1;
            if (t < 32) {
                if (k + 1 < total) {
                    const int kn  = k + 1;
                    const int row = rbeg + kn / nchunks;
                    const int nc0 = (kn % nchunks) * CHUNK_PX;
                    int npx = w - nc0; if (npx > CHUNK_PX) npx = CHUNK_PX;
                    tdm_load_row_chunk(fb + ((size_t)row * FW + x + nc0) * FC,
                                       cur ? lds0 : lds1, npx * FC);
                    // chunk k complete (TDM in-order per wave), k+1 in flight
                    __builtin_amdgcn_s_wait_tensorcnt(1);
                } else {
                    __builtin_amdgcn_s_wait_tensorcnt(0);
                }
            }
            __syncthreads();   // publish chunk k's LDS data to all waves

            const int c0  = (k % nchunks) * CHUNK_PX;
            int npx = w - c0; if (npx > CHUNK_PX) npx = CHUNK_PX;
            const float* buf = &sbuf[cur][0];
#pragma unroll 4
            for (int p = 0; p < npx; ++p) {
                float v = buf[p * FC + t];           // ds_load_b32, bank-friendly
                int col = c0 + p;                    // column relative to x
                // ccell = clip(col / dw, 0, 2) via thresholds
                if (col < dw)            m0 = fmaxf(m0, v);
                else if (col < 2 * dw)   m1 = fmaxf(m1, v);
                else                     m2 = fmaxf(m2, v);
            }
            __syncthreads();   // all reads done before this buffer is re-targeted
        }

        // pooled layout: (B, NREG, PH, PW, C)
        size_t obase = (((size_t)blk * PH + rc) * PW) * FC + t;
        pooled[obase + 0 * FC] = m0;
        pooled[obase + 1 * FC] = m1;
        pooled[obase + 2 * FC] = m2;
    }
}

// ---------------------------------------------------------------------------
extern "C" void kernel_launch(void* const* d_in, const int* in_sizes, int n_in,
                              void* d_out, int out_size, void* d_ws, size_t ws_size,
                              hipStream_t stream) {
    const float* features = (const float*)d_in[0];           // (4,64,64,256) f32
    const float* roi      = (const float*)d_in[1];           // (4,128,4) f32

    float* pooled  = (float*)d_out;                          // (4,32,3,3,256)
    float* roi_out = pooled + (size_t)BATCH * NREG * PH * PW * FC; // (4,32,4)
    int*   boxes   = (int*)d_ws;                             // (4,32,4) int scratch

    nms_clip_kernel<<<BATCH, RBOX, 0, stream>>>(roi, boxes, roi_out);
    pool_kernel<<<BATCH * NREG, FC, 0, stream>>>(features, boxes, pooled);
}